// GraphSage_23888608101401
// MI455X (gfx1250) — compile-verified
//
#include <hip/hip_runtime.h>

typedef __attribute__((ext_vector_type(2))) float v2f;
typedef __attribute__((ext_vector_type(8))) float v8f;

#define EMB 32
#define K_NBR 10
#define WAVES_PER_BLOCK 8

// One wave32 computes a 16-row tile of:  relu(concat(self, mean(neigh)) @ W^T)
// M=16 rows, K=64 (concat dim), N=32 output features.
// WMMA f32 16x16x4: 16 K-steps x 2 N-halves = 32 v_wmma per tile.
__global__ __launch_bounds__(256) void sage_layer_kernel(
    const float* __restrict__ feat,    // [*, 32] feature source (emb or h1)
    const float* __restrict__ W,       // [32, 64] row-major (out x in)
    const int*   __restrict__ neigh,   // [N_NODES, 10]
    const int*   __restrict__ idxMap,  // [rows] node ids, or nullptr (identity)
    float*       __restrict__ out,     // [rows, 32]
    int nTiles)
{
    __shared__ float cshAll[WAVES_PER_BLOCK * 16 * 64];   // 32 KB: combined[16][64] per wave
    const int lane   = threadIdx.x & 31;
    const int wave   = threadIdx.x >> 5;
    const int tile   = blockIdx.x * WAVES_PER_BLOCK + wave;
    if (tile >= nTiles) return;                            // uniform per wave; EXEC stays all-1s

    const int half   = lane >> 4;    // 0: K pair {0,1}; 1: K pair {2,3} of each 4-wide K step
    const int lane16 = lane & 15;
    float* csh = &cshAll[wave * 16 * 64];

    // ---- Preload B fragments: B[k][n] = W[n][k]  (16x16x4 fp32 B layout) ----
    // VGPR0 = K {0 | 2}, VGPR1 = K {1 | 3} across the two lane halves; N = lane16 (+16 per half).
    v2f Bf[2][16];
    #pragma unroll
    for (int nh = 0; nh < 2; ++nh) {
        const int o = lane16 + 16 * nh;                    // output feature (row of W)
        #pragma unroll
        for (int k = 0; k < 16; ++k) {
            const int kk = 4 * k + 2 * half;
            const float2 w2 = *(const float2*)(W + o * 64 + kk);
            v2f b; b.x = w2.x; b.y = w2.y;
            Bf[nh][k] = b;
        }
    }

    // ---- Gather phase: build combined[16][64] in LDS ----
    // 32 lanes <-> 32 features: every feature-row read is one coalesced 128B line (L2-resident).
    const int m0 = tile * 16;
    for (int r = 0; r < 16; ++r) {
        const int row  = m0 + r;
        const int node = idxMap ? idxMap[row] : row;
        const float self = feat[node * EMB + lane];
        const int* nb = neigh + node * K_NBR;
        float acc = 0.f;
        #pragma unroll
        for (int j = 0; j < K_NBR; ++j)
            acc += feat[nb[j] * EMB + lane];
        csh[r * 64 + lane]      = self;                    // K = 0..31  (self features)
        csh[r * 64 + 32 + lane] = acc * (1.0f / K_NBR);    // K = 32..63 (mean aggregate)
    }

    // ---- WMMA phase: D[16x16] per N-half, accumulate over 16 K-steps ----
    #pragma unroll
    for (int nh = 0; nh < 2; ++nh) {
        v8f c = {};
        #pragma unroll
        for (int k = 0; k < 16; ++k) {
            // A fragment: row = lane16, K pair (4k + 2*half, +1)  -> one ds_load_b64
            const float2 a2 = *(const float2*)&csh[lane16 * 64 + 4 * k + 2 * half];
            v2f a; a.x = a2.x; a.y = a2.y;
            c = __builtin_amdgcn_wmma_f32_16x16x4_f32(
                    /*neg_a=*/false, a, /*neg_b=*/false, Bf[nh][k],
                    /*c_mod=*/(short)0, c, /*reuse_a=*/false, /*reuse_b=*/false);
        }
        // C/D layout: VGPR v -> M = v + 8*half, N = lane16
        const int o = 16 * nh + lane16;
        #pragma unroll
        for (int v = 0; v < 8; ++v) {
            const int m = v + 8 * half;
            out[(m0 + m) * EMB + o] = fmaxf(c[v], 0.0f);   // fused ReLU
        }
    }
}

extern "C" void kernel_launch(void* const* d_in, const int* in_sizes, int n_in,
                              void* d_out, int out_size, void* d_ws, size_t ws_size,
                              hipStream_t stream) {
    const float* emb        = (const float*)d_in[0];   // [500000, 32]
    const float* W1         = (const float*)d_in[1];   // [32, 64]
    const float* W2         = (const float*)d_in[2];   // [32, 64]
    const int*   node_batch = (const int*)  d_in[3];   // [100000]
    const int*   neigh      = (const int*)  d_in[4];   // [500000, 10]
    float*       out        = (float*)d_out;           // [100000, 32]
    float*       h1         = (float*)d_ws;            // [500000, 32] = 64 MB scratch

    const int N = in_sizes[0] / EMB;    // 500000 (divisible by 16)
    const int B = in_sizes[3];          // 100000 (divisible by 16)

    const int t1 = (N + 15) / 16;
    const int t2 = (B + 15) / 16;

    // Layer 1: h1 = relu(concat(emb, mean(emb[neigh])) @ W1^T) for all nodes
    sage_layer_kernel<<<(t1 + WAVES_PER_BLOCK - 1) / WAVES_PER_BLOCK, 256, 0, stream>>>(
        emb, W1, neigh, nullptr, h1, t1);

    // Layer 2: out = relu(concat(h1[batch], mean(h1[neigh[batch]])) @ W2^T)
    sage_layer_kernel<<<(t2 + WAVES_PER_BLOCK - 1) / WAVES_PER_BLOCK, 256, 0, stream>>>(
        h1, W2, neigh, node_batch, out, t2);
}